// mHC_36112085025013
// MI455X (gfx1250) — compile-verified
//
#include <hip/hip_runtime.h>
#include <hip/hip_bf16.h>
#include <math.h>

// ---------------------------------------------------------------------------
// mHC fused kernel for MI455X (gfx1250, wave32)
//   K1: fused post_mapping + sumsq + raw logits GEMM (WMMA f32 16x16x4)
//       + async double-buffered fn-tile staging (GLOBAL_LOAD_ASYNC_TO_LDS)
//       + per-row softmax/sigmoid/Sinkhorn epilogue
//   K2: layer_input = sum_n h_pre[n] * residual_cur[:,n,:]
// ---------------------------------------------------------------------------

typedef __attribute__((ext_vector_type(2))) float v2f;
typedef __attribute__((ext_vector_type(8))) float v8f;

#define BQ    4096
#define HID   4096
#define NMULT 4
#define HC    16384          // NMULT * HID
#define MIXHC 24
#define RB    16             // batch rows per block == WMMA M
#define KB    128            // flat-d chunk per iteration
#define LPAD  4
#define LDA   (KB + LPAD)    // 132 floats: (132 % 64) == 4 -> conflict-free cols
#define SB_STRIDE (32 * LDA) // one fn-tile buffer

__device__ __forceinline__ float4 f4fma(float s, float4 a, float4 b) {
    return make_float4(fmaf(s, a.x, b.x), fmaf(s, a.y, b.y),
                       fmaf(s, a.z, b.z), fmaf(s, a.w, b.w));
}

// Issue async global->LDS copies of one fn tile (rows 0..23, KB columns).
// 24*128 floats = 768 b128 transfers = 3 per thread (256 threads).
__device__ __forceinline__ void stage_fn_async(const float* __restrict__ fn,
                                               float* s_b_buf, int kb, int tid) {
#pragma unroll
    for (int v = 0; v < 3; ++v) {
        const int idx = tid + v * 256;          // 0..767
        const int m   = idx >> 5;               // 0..23
        const int c   = (idx & 31) * 4;         // 0..124
        const unsigned long long g =
            (unsigned long long)(size_t)(fn + (long)m * HC + kb + c);
        const unsigned l = (unsigned)(size_t)(s_b_buf + m * LDA + c);
        asm volatile("global_load_async_to_lds_b128 %0, %1, off"
                     :: "v"(l), "v"(g) : "memory");
    }
}

__global__ __launch_bounds__(256) void mhc_fused_kernel(
    const float* __restrict__ x_prev,      // [B, HID]
    const float* __restrict__ r_prev,      // [B, 4, HID]
    const float* __restrict__ post_prev,   // [B, 4]
    const float* __restrict__ comb_prev,   // [B, 4, 4]
    const float* __restrict__ fn,          // [24, HC]
    const float* __restrict__ base,        // [24]
    const float* __restrict__ scale,       // [3]
    float* __restrict__ out_rcur,          // [B, 4, HID]
    float* __restrict__ out_post,          // [B, 4]
    float* __restrict__ out_comb,          // [B, 4, 4]
    float* __restrict__ hpre_ws)           // [B, 4] workspace
{
    __shared__ float s_a[RB * LDA];        // residual_cur tile (A matrix)
    __shared__ float s_b[2 * SB_STRIDE];   // fn tiles (double buffered),
                                           // rows 24..31 zero-padded once
    __shared__ float s_comb[RB * 16];
    __shared__ float s_post[RB * 4];
    __shared__ float s_logits[RB * 32];    // C accumulation across waves
    __shared__ float s_red[256];
    __shared__ float s_rms[RB];

    const int  tid  = threadIdx.x;
    const int  wave = tid >> 5;
    const int  lane = tid & 31;
    const int  half = lane >> 4;           // K half (0: k0,k1 ; 1: k2,k3)
    const int  lrow = lane & 15;           // M for A-frag, N for B-frag
    const long b0   = (long)blockIdx.x * RB;

    // stage per-row mixing params; zero logits accumulator + fn pad rows
    s_comb[tid] = comb_prev[b0 * 16 + tid];               // 256 elements
    if (tid < RB * 4) s_post[tid] = post_prev[b0 * 4 + tid];
    s_logits[tid]       = 0.f;
    s_logits[tid + 256] = 0.f;
#pragma unroll
    for (int z = 0; z < 8; ++z) {          // 2 bufs * 8 pad rows * 128 cols
        const int e    = tid + z * 256;
        const int bsel = e >> 10;          // /(8*KB)
        const int rr   = (e >> 7) & 7;     // pad row 0..7 -> m = 24+rr
        const int cc   = e & 127;
        s_b[bsel * SB_STRIDE + (24 + rr) * LDA + cc] = 0.f;
    }
    // kick off async staging of the first fn tile into buffer 0
    stage_fn_async(fn, s_b, 0, tid);
    __syncthreads();

    // phase-1 thread mapping: 16 threads per batch row, 8 contiguous d each
    const int  prow  = tid >> 4;
    const int  dl    = (tid & 15) * 8;
    const long row_g = b0 + prow;

    float sumsq = 0.f;
    v8f c0 = {0.f,0.f,0.f,0.f,0.f,0.f,0.f,0.f};
    v8f c1 = {0.f,0.f,0.f,0.f,0.f,0.f,0.f,0.f};
    int buf = 0;

    for (int i_str = 0; i_str < NMULT; ++i_str) {
        // hoisted mixing coefficients: constant over the inner loop
        const float cw0 = s_comb[prow * 16 + i_str * 4 + 0];
        const float cw1 = s_comb[prow * 16 + i_str * 4 + 1];
        const float cw2 = s_comb[prow * 16 + i_str * 4 + 2];
        const float cw3 = s_comb[prow * 16 + i_str * 4 + 3];
        const float pw  = s_post[prow * 4 + i_str];
        const float* rb0 = r_prev + row_g * (NMULT * HID);
        float*       ob  = out_rcur + row_g * (NMULT * HID) + i_str * HID;

        for (int hb = 0; hb < HID; hb += KB) {
            const int kb = i_str * HID + hb;   // flat hyper-connection offset
            const int h  = hb + dl;

            // ---- phase 1: stream in, mix, write out, stage A tile ----
            {
                const float4* xp  = (const float4*)(x_prev + row_g * HID + h);
                const float4* rp0 = (const float4*)(rb0 + 0 * HID + h);
                const float4* rp1 = (const float4*)(rb0 + 1 * HID + h);
                const float4* rp2 = (const float4*)(rb0 + 2 * HID + h);
                const float4* rp3 = (const float4*)(rb0 + 3 * HID + h);

                // prefetch next chunk's streams (global_prefetch_b8)
                if (kb + KB < HC) {
                    const int nd = kb + KB;
                    const int ni = nd >> 12;
                    const int nh = (nd & (HID - 1)) + dl;
                    __builtin_prefetch(rb0 + ni * HID + nh, 0, 3);
                    __builtin_prefetch(x_prev + row_g * HID + nh, 0, 3);
                }

#pragma unroll
                for (int v = 0; v < 2; ++v) {
                    float4 acc = xp[v];
                    acc = make_float4(pw*acc.x, pw*acc.y, pw*acc.z, pw*acc.w);
                    acc = f4fma(cw0, rp0[v], acc);
                    acc = f4fma(cw1, rp1[v], acc);
                    acc = f4fma(cw2, rp2[v], acc);
                    acc = f4fma(cw3, rp3[v], acc);
                    ((float4*)(ob + h))[v] = acc;
                    const int o = prow * LDA + dl + v * 4;
                    s_a[o + 0] = acc.x; s_a[o + 1] = acc.y;
                    s_a[o + 2] = acc.z; s_a[o + 3] = acc.w;
                    sumsq = fmaf(acc.x, acc.x, sumsq);
                    sumsq = fmaf(acc.y, acc.y, sumsq);
                    sumsq = fmaf(acc.z, acc.z, sumsq);
                    sumsq = fmaf(acc.w, acc.w, sumsq);
                }
            }

            // ---- async pipeline: issue next fn tile, wait for current ----
            if (kb + KB < HC) {
                stage_fn_async(fn, s_b + (buf ^ 1) * SB_STRIDE, kb + KB, tid);
                // current tile (3 older ops) complete; next 3 may remain in flight
                asm volatile("s_wait_asynccnt 0x3" ::: "memory");
            } else {
                asm volatile("s_wait_asynccnt 0x0" ::: "memory");
            }
            __syncthreads();

            // ---- phase 2: rawdot += A_tile @ fn_tile (WMMA f32 16x16x4) ----
            // wave w owns k-steps [4w, 4w+4); each k-step covers 4 flat d.
            const float* sb = s_b + buf * SB_STRIDE;
#pragma unroll
            for (int q = 0; q < 4; ++q) {
                const int dlk = (wave * 4 + q) * 4 + 2 * half;
                v2f a, bA, bB;
                a.x  = s_a[lrow * LDA + dlk];        // A[M=lrow][2*half(+1)]
                a.y  = s_a[lrow * LDA + dlk + 1];
                bA.x = sb[lrow * LDA + dlk];         // B[k][N], tile m=0..15
                bA.y = sb[lrow * LDA + dlk + 1];
                bB.x = sb[(16 + lrow) * LDA + dlk];  // tile m=16..31 (zeros >=24)
                bB.y = sb[(16 + lrow) * LDA + dlk + 1];
                c0 = __builtin_amdgcn_wmma_f32_16x16x4_f32(
                         false, a, false, bA, (short)0, c0, false, false);
                c1 = __builtin_amdgcn_wmma_f32_16x16x4_f32(
                         false, a, false, bB, (short)0, c1, false, false);
            }
            __syncthreads();
            buf ^= 1;
        }
    }

    // ---------- reductions ----------
    s_red[tid] = sumsq;                    // tid == prow*16 + seg
    __syncthreads();
    if (tid < RB) {
        float s = 0.f;
#pragma unroll
        for (int e = 0; e < 16; ++e) s += s_red[tid * 16 + e];
        s_rms[tid] = sqrtf(s * (1.0f / (float)HC) + 1e-6f);
    }
    // merge per-wave WMMA accumulators: C[M][N], M = r + 8*half, N = lrow
#pragma unroll
    for (int r = 0; r < 8; ++r) {
        const int M = r + 8 * half;
        atomicAdd(&s_logits[M * 32 + lrow],      c0[r]);
        atomicAdd(&s_logits[M * 32 + 16 + lrow], c1[r]);
    }
    __syncthreads();

    // ---------- per-row epilogue: softmax / sigmoid / Sinkhorn ----------
    if (tid < RB) {
        const int  r = tid;
        const long b = b0 + r;
        const float inv_rms = 1.0f / s_rms[r];     // logits = rawdot / rms
        float lg[MIXHC];
#pragma unroll
        for (int m = 0; m < MIXHC; ++m) lg[m] = s_logits[r * 32 + m] * inv_rms;

        const float sc0 = scale[0], sc1 = scale[1], sc2 = scale[2];

        // h_pre = softmax(lg[0:4]*sc0 + base[0:4])
        float pre[4], mx = -1e30f;
#pragma unroll
        for (int i = 0; i < 4; ++i) {
            pre[i] = fmaf(lg[i], sc0, base[i]);
            mx = fmaxf(mx, pre[i]);
        }
        float hp[4], es = 0.f;
#pragma unroll
        for (int i = 0; i < 4; ++i) { hp[i] = __expf(pre[i] - mx); es += hp[i]; }
        const float ies = 1.0f / es;
#pragma unroll
        for (int i = 0; i < 4; ++i) { hp[i] *= ies; hpre_ws[b * 4 + i] = hp[i]; }

        // post_mix = sigmoid(lg[4:8]*sc1 + base[4:8])   (POST_MULT == 1)
#pragma unroll
        for (int i = 0; i < 4; ++i) {
            const float p = fmaf(lg[4 + i], sc1, base[4 + i]);
            out_post[b * 4 + i] = 1.0f / (1.0f + __expf(-p));
        }

        // comb_mix = Sinkhorn(exp(lg[8:24]*sc2 + base[8:24]), 10 iters)
        float Mm[4][4];
#pragma unroll
        for (int i = 0; i < 4; ++i)
#pragma unroll
            for (int j = 0; j < 4; ++j)
                Mm[i][j] = __expf(fmaf(lg[8 + i*4 + j], sc2, base[8 + i*4 + j]));
        for (int it = 0; it < 10; ++it) {
#pragma unroll
            for (int i = 0; i < 4; ++i) {
                const float rs = 1.0f / (Mm[i][0]+Mm[i][1]+Mm[i][2]+Mm[i][3] + 1e-6f);
#pragma unroll
                for (int j = 0; j < 4; ++j) Mm[i][j] *= rs;
            }
#pragma unroll
            for (int j = 0; j < 4; ++j) {
                const float cs = 1.0f / (Mm[0][j]+Mm[1][j]+Mm[2][j]+Mm[3][j] + 1e-6f);
#pragma unroll
                for (int i = 0; i < 4; ++i) Mm[i][j] *= cs;
            }
        }
#pragma unroll
        for (int i = 0; i < 4; ++i)
#pragma unroll
            for (int j = 0; j < 4; ++j)
                out_comb[b * 16 + i * 4 + j] = Mm[i][j];
    }
}

// layer_input[b,h] = sum_n h_pre[b,n] * residual_cur[b,n,h]
__global__ __launch_bounds__(256) void mhc_layer_input_kernel(
    const float* __restrict__ rcur,   // [B, 4, HID]
    const float* __restrict__ hpre,   // [B, 4]
    float* __restrict__ out_layer)    // [B, HID]
{
    const long b = blockIdx.x;
    const float h0 = hpre[b * 4 + 0];
    const float h1 = hpre[b * 4 + 1];
    const float h2 = hpre[b * 4 + 2];
    const float h3 = hpre[b * 4 + 3];
    const float* rb = rcur + b * (NMULT * HID);
    float* ob = out_layer + b * HID;

    for (int t = threadIdx.x; t < HID / 4; t += 256) {
        const float4 r0 = ((const float4*)(rb + 0 * HID))[t];
        const float4 r1 = ((const float4*)(rb + 1 * HID))[t];
        const float4 r2 = ((const float4*)(rb + 2 * HID))[t];
        const float4 r3 = ((const float4*)(rb + 3 * HID))[t];
        float4 o;
        o.x = fmaf(h3, r3.x, fmaf(h2, r2.x, fmaf(h1, r1.x, h0 * r0.x)));
        o.y = fmaf(h3, r3.y, fmaf(h2, r2.y, fmaf(h1, r1.y, h0 * r0.y)));
        o.z = fmaf(h3, r3.z, fmaf(h2, r2.z, fmaf(h1, r1.z, h0 * r0.z)));
        o.w = fmaf(h3, r3.w, fmaf(h2, r2.w, fmaf(h1, r1.w, h0 * r0.w)));
        ((float4*)ob)[t] = o;
    }
}

extern "C" void kernel_launch(void* const* d_in, const int* in_sizes, int n_in,
                              void* d_out, int out_size, void* d_ws, size_t ws_size,
                              hipStream_t stream) {
    const float* x_prev    = (const float*)d_in[0];
    const float* r_prev    = (const float*)d_in[1];
    const float* post_prev = (const float*)d_in[2];
    const float* comb_prev = (const float*)d_in[3];
    const float* fn        = (const float*)d_in[4];
    const float* base      = (const float*)d_in[5];
    const float* scale     = (const float*)d_in[6];

    float* out       = (float*)d_out;
    float* out_rcur  = out;                                        // B*4*HID
    float* out_post  = out_rcur + (size_t)BQ * NMULT * HID;        // B*4
    float* out_comb  = out_post + (size_t)BQ * NMULT;              // B*4*4
    float* out_layer = out_comb + (size_t)BQ * NMULT * NMULT;      // B*HID
    float* hpre_ws   = (float*)d_ws;                               // B*4 floats

    mhc_fused_kernel<<<dim3(BQ / RB), dim3(256), 0, stream>>>(
        x_prev, r_prev, post_prev, comb_prev, fn, base, scale,
        out_rcur, out_post, out_comb, hpre_ws);

    mhc_layer_input_kernel<<<dim3(BQ), dim3(256), 0, stream>>>(
        out_rcur, hpre_ws, out_layer);
}